// lossFunc_36953898615431
// MI455X (gfx1250) — compile-verified
//
#include <hip/hip_runtime.h>
#include <hip/hip_bf16.h>

#define BB 128
#define TT 200
#define QQ 4096
#define MM (TT - 1)   // 199

typedef float __attribute__((ext_vector_type(4))) f4;
typedef float __attribute__((ext_vector_type(2))) v2f;
typedef float __attribute__((ext_vector_type(8))) v8f;

// ---------------------------------------------------------------------------
// Kernel 1: locate the one-hot in each [2Q] row of `batch`.
// One block per (b,t) row: 256 threads x 8 coalesced b128 NT loads = 32 KB.
// Encode position as sum((i+1)*x) -> exact in f32 (max 8192).
// ---------------------------------------------------------------------------
__global__ __launch_bounds__(256) void scan_onehot(const float* __restrict__ batch,
                                                   int* __restrict__ qsel) {
    const size_t row = blockIdx.x;                  // 0 .. B*T-1
    const f4* base = (const f4*)(batch + row * (size_t)(2 * QQ));
    const int tid = threadIdx.x;

    float idxSum = 0.0f;
#pragma unroll
    for (int k = 0; k < 8; ++k) {
        // flat f4 index: consecutive lanes -> consecutive 16B -> fully coalesced
        const int f = tid + k * 256;
        f4 v = __builtin_nontemporal_load(&base[f]);
        const float i0 = (float)(4 * f);
        idxSum += v.x * (i0 + 1.0f) + v.y * (i0 + 2.0f) +
                  v.z * (i0 + 3.0f) + v.w * (i0 + 4.0f);
    }

    // wave32 reduce
#pragma unroll
    for (int m = 1; m < 32; m <<= 1) idxSum += __shfl_xor(idxSum, m, 32);

    __shared__ float red[8];
    if ((tid & 31) == 0) red[tid >> 5] = idxSum;
    __syncthreads();
    if (tid == 0) {
        float s = 0.0f;
#pragma unroll
        for (int w = 0; w < 8; ++w) s += red[w];
        qsel[row] = (int)s - 1;                     // 0 .. 8191
    }
}

// ---------------------------------------------------------------------------
// Kernel 2: per-student trim + BCE. One block per student, thread t handles
// timestep t (t < 199). Writes p*mask, a*mask, mask into d_out and the
// per-student mean into workspace.
// ---------------------------------------------------------------------------
__global__ __launch_bounds__(256) void per_student(const float* __restrict__ pred,
                                                   const int* __restrict__ qsel,
                                                   float* __restrict__ out,
                                                   float* __restrict__ psum) {
    const int b = blockIdx.x;
    const int t = threadIdx.x;
    const bool act = (t < MM);

    float p = 0.0f, a = 0.0f;
    if (act) {
        const int idx = qsel[b * TT + t + 1];
        const int q = idx & (QQ - 1);
        a = (idx < QQ) ? 1.0f : 0.0f;
        p = pred[((size_t)b * TT + t) * QQ + q];
    }

    // ---- block max of last index with p > 0 (or -1) ----
    int v = (act && p > 0.0f) ? t : -1;
#pragma unroll
    for (int m = 1; m < 32; m <<= 1) v = max(v, __shfl_xor(v, m, 32));
    __shared__ int ired[8];
    if ((t & 31) == 0) ired[t >> 5] = v;
    __syncthreads();
    int last = ired[0];
#pragma unroll
    for (int w = 1; w < 8; ++w) last = max(last, ired[w]);
    if (last < 0) last = MM - 1;                    // no p>0 -> full mask

    const float mask = (act && t <= last) ? 1.0f : 0.0f;

    // BCE with torch-style log clamp at -100 (clamp also kills inf for p==0)
    const float logp   = fmaxf(logf(p), -100.0f);
    const float log1mp = fmaxf(log1pf(-p), -100.0f);
    const float bce = -(a * logp + (1.0f - a) * log1mp);

    float num = bce * mask;
    float den = mask;
#pragma unroll
    for (int m = 1; m < 32; m <<= 1) {
        num += __shfl_xor(num, m, 32);
        den += __shfl_xor(den, m, 32);
    }
    __shared__ float fnum[8], fden[8];
    if ((t & 31) == 0) { fnum[t >> 5] = num; fden[t >> 5] = den; }
    __syncthreads();
    if (t == 0) {
        float ns = 0.0f, ds = 0.0f;
#pragma unroll
        for (int w = 0; w < 8; ++w) { ns += fnum[w]; ds += fden[w]; }
        psum[b] = ns / ds;
    }

    if (act) {
        const size_t off = (size_t)b * MM + t;
        out[1 + off]                        = p * mask;
        out[1 + (size_t)BB * MM + off]      = a * mask;
        out[1 + (size_t)2 * BB * MM + off]  = mask;
    }
}

// ---------------------------------------------------------------------------
// Kernel 3: loss = sum of 128 per-student means, via V_WMMA_F32_16X16X4_F32.
// A = ones(16x4), B = values(4x16): D[0][n] = sum_k B[k][n] (exact f32 FMA).
// Two WMMAs (64 values each) accumulate into C; then 4 lane-xor adds finish.
// Single wave32, full EXEC (required for WMMA).
// ---------------------------------------------------------------------------
__global__ __launch_bounds__(32) void loss_sum_wmma(const float* __restrict__ psum,
                                                    float* __restrict__ out) {
    const int lane = threadIdx.x;                   // 0..31
    const int n = lane & 15;
    const int hi = (lane >= 16) ? 1 : 0;

    // B-matrix (4x16) f32 layout: VGPR0 lanes 0-15 = B[0][n], lanes 16-31 = B[2][n]
    //                             VGPR1 lanes 0-15 = B[1][n], lanes 16-31 = B[3][n]
    v2f b0, b1, ones;
    b0.x = psum[(hi ? 2 : 0) * 16 + n];
    b0.y = psum[(hi ? 3 : 1) * 16 + n];
    b1.x = psum[64 + (hi ? 2 : 0) * 16 + n];
    b1.y = psum[64 + (hi ? 3 : 1) * 16 + n];
    ones.x = 1.0f; ones.y = 1.0f;

    v8f c = {};
    c = __builtin_amdgcn_wmma_f32_16x16x4_f32(false, ones, false, b0, (short)0, c,
                                              false, false);
    c = __builtin_amdgcn_wmma_f32_16x16x4_f32(false, ones, false, b1, (short)0, c,
                                              false, false);

    // c[0]: lanes 0-15 hold D[0][n] = column sums; reduce 16 lanes.
    float s = c[0];
    s += __shfl_xor(s, 1, 32);
    s += __shfl_xor(s, 2, 32);
    s += __shfl_xor(s, 4, 32);
    s += __shfl_xor(s, 8, 32);
    if (lane == 0) out[0] = s;
}

// ---------------------------------------------------------------------------
extern "C" void kernel_launch(void* const* d_in, const int* in_sizes, int n_in,
                              void* d_out, int out_size, void* d_ws, size_t ws_size,
                              hipStream_t stream) {
    const float* pred  = (const float*)d_in[0];   // [B,T,Q] f32
    const float* batch = (const float*)d_in[1];   // [B,T,2Q] f32
    float* out = (float*)d_out;                   // [1 + 3*B*(T-1)] f32

    int*   qsel = (int*)d_ws;                                          // B*T ints
    float* psum = (float*)((char*)d_ws + (size_t)BB * TT * sizeof(int)); // B floats

    scan_onehot<<<BB * TT, 256, 0, stream>>>(batch, qsel);
    per_student<<<BB, 256, 0, stream>>>(pred, qsel, out, psum);
    loss_sum_wmma<<<1, 32, 0, stream>>>(psum, out);
}